// RNNClassifier_78537771975213
// MI455X (gfx1250) — compile-verified
//
#include <hip/hip_runtime.h>

// ---------------------------------------------------------------------------
// Fused LSTM (B=4096, T=2048, D=10, H=4) + softmax(30) head for gfx1250.
//
// One wave32 = 16 batch rows. Per timestep, gates[16x16] come from a chain of
// 4 x V_WMMA_F32_16X16X4_F32:  acc = bias; acc += X(16x12pad)*W_ih^T (3 K-slices,
// depend only on loads); acc += h(16x4)*W_hh^T (last, the only op on the
// recurrent serial chain). Activations computed in C-layout; transposed to
// per-unit layout via a per-wave LDS tile (stride 20 dwords, conflict-free);
// the reloads come back exactly in WMMA A-layout so h feeds the next WMMA.
//
// X loads are software-pipelined 2 iterations deep (plus global_prefetch_b8
// ~24 steps ahead) because with 256 waves there is <=1 wave/SIMD: all latency
// must be hidden inside the wave's own instruction stream.
// ---------------------------------------------------------------------------

typedef __attribute__((ext_vector_type(2))) float v2f;
typedef __attribute__((ext_vector_type(8))) float v8f;

#define TT 2048   // timesteps
#define TD 10     // input dim
#define THH 4     // hidden units
#define TOUT 30   // output classes
#define RS 20     // LDS row stride (dwords); halves of the wave hit disjoint banks
#define LOG2E 1.4426950408889634f

__device__ __forceinline__ float tanh_fast(float x) {
  // tanh(x) = 1 - 2/(exp(2x)+1)
  float e = __builtin_amdgcn_exp2f(x * (2.0f * LOG2E));
  return fmaf(-2.0f, __builtin_amdgcn_rcpf(e + 1.0f), 1.0f);
}

__global__ __launch_bounds__(32) void lstm_wmma_scan(
    const float* __restrict__ X,      // [B, T, D]
    const float* __restrict__ W_ih,   // [16, 10]  gate order i,f,g,o
    const float* __restrict__ W_hh,   // [16, 4]
    const float* __restrict__ b_ih,   // [16]
    const float* __restrict__ b_hh,   // [16]
    const float* __restrict__ W_out,  // [30, 4]
    const float* __restrict__ b_out,  // [30]
    float* __restrict__ out) {        // [B, 30] softmax probs
  __shared__ float sw[16 * RS];

  const int lane = threadIdx.x & 31;
  const int hiH  = lane >> 4;     // wave half: 0 or 1
  const int n    = lane & 15;     // gate column (C layout) / batch row (A layout)
  const int kb   = hiH * 2;       // K base carried by this half (A/B layout)
  const int b0   = blockIdx.x * 16;

  // --- B operands (4x16, f32): VGPR0 = K{kb}, VGPR1 = K{kb+1}, lane = N ---
  v2f bh  = { W_hh[n * THH + kb], W_hh[n * THH + kb + 1] };          // W_hh^T
  v2f bx0 = { W_ih[n * TD + kb],          W_ih[n * TD + kb + 1] };   // K 0..3
  v2f bx1 = { W_ih[n * TD + 4 + kb],      W_ih[n * TD + 4 + kb + 1]};// K 4..7
  v2f bx2;                                                           // K 8..11 (10,11 -> 0)
  bx2.x = (8 + kb     < TD) ? W_ih[n * TD + 8 + kb]     : 0.0f;
  bx2.y = (8 + kb + 1 < TD) ? W_ih[n * TD + 8 + kb + 1] : 0.0f;

  const float biasL = b_ih[n] + b_hh[n];
  v8f bias8 = { biasL, biasL, biasL, biasL, biasL, biasL, biasL, biasL };

  // Activation selectors: columns 8..11 are the g-gate -> tanh = 2*sig(2x)-1
  const bool  isg    = (n >= 8) && (n < 12);
  const float negPre = (isg ? -2.0f : -1.0f) * LOG2E; // feeds exp2
  const float postA  = isg ? 2.0f : 1.0f;
  const float postB  = isg ? -1.0f : 0.0f;

  // --- X streams in A-layout: lane owns row m=n, K slice kb.. ---
  const float* pr  = X + (size_t)(b0 + n) * (size_t)(TT * TD);
  const float* pA0 = pr + kb;       // chunk0: K = kb, kb+1
  const float* pA1 = pr + 4 + kb;   // chunk1
  const float* pA2 = pr + 8;        // chunk2: K=8,9 both halves (B zeroed past D)

  // LDS addressing: store act[r] at row (r + 8*half), col n;
  // reload i/f/g/o for (row m=n, units kb,kb+1) -> exactly next A-layout h.
  float*    swi  = sw + n * RS + kb;
  const int srow = hiH ? 8 : 0;

  v2f   hA = {0.0f, 0.0f};  // h in A-layout (this lane's 2 units of row n)
  float c0 = 0.0f, c1 = 0.0f;

  // Software pipeline: cur = data for t, nxt = data for t+1; issue t+2 early.
  v2f a0c = *(const v2f*)(pA0);
  v2f a1c = *(const v2f*)(pA1);
  v2f a2c = *(const v2f*)(pA2);
  v2f a0n = *(const v2f*)(pA0 + TD);
  v2f a1n = *(const v2f*)(pA1 + TD);
  v2f a2n = *(const v2f*)(pA2 + TD);

  for (int t = 0; t < TT; ++t) {
    v2f ax0 = a0c, ax1 = a1c, ax2 = a2c;
    a0c = a0n; a1c = a1n; a2c = a2n;

    // Issue loads for t+2 immediately (clamped; ~2 loop bodies before use).
    {
      int tn = t + 2; if (tn > TT - 1) tn = TT - 1;
      const size_t offn = (size_t)tn * TD;
      a0n = *(const v2f*)(pA0 + offn);
      a1n = *(const v2f*)(pA1 + offn);
      a2n = *(const v2f*)(pA2 + offn);
      __builtin_prefetch(pA0 + offn + 24 * TD, 0, 0);  // warm GL2 ~1KB ahead
    }

    // gates = bias + x*W_ih^T (load-dependent, issue first)
    //               + h*W_hh^T (recurrent-dependent, issue last)
    v8f acc = __builtin_amdgcn_wmma_f32_16x16x4_f32(false, ax0, false, bx0,
                                                    (short)0, bias8, false, false);
    acc = __builtin_amdgcn_wmma_f32_16x16x4_f32(false, ax1, false, bx1,
                                                (short)0, acc, false, false);
    acc = __builtin_amdgcn_wmma_f32_16x16x4_f32(false, ax2, false, bx2,
                                                (short)0, acc, false, false);
    acc = __builtin_amdgcn_wmma_f32_16x16x4_f32(false, hA,  false, bh,
                                                (short)0, acc, false, false);

    // per-column activation (sigmoid, or tanh for g columns) + LDS transpose
#pragma unroll
    for (int r = 0; r < 8; ++r) {
      float s = __builtin_amdgcn_rcpf(
          1.0f + __builtin_amdgcn_exp2f(acc[r] * negPre));
      sw[(r + srow) * RS + n] = fmaf(s, postA, postB);
    }
    // reload in per-unit layout (in-order DS within the single wave of this block)
    v2f iv = *(const v2f*)(swi);
    v2f fv = *(const v2f*)(swi + 4);
    v2f gv = *(const v2f*)(swi + 8);
    v2f ov = *(const v2f*)(swi + 12);

    c0 = fmaf(fv.x, c0, iv.x * gv.x);
    c1 = fmaf(fv.y, c1, iv.y * gv.y);
    hA.x = ov.x * tanh_fast(c0);
    hA.y = ov.y * tanh_fast(c1);
  }

  // ---------------- softmax head, fused (16 rows per wave) ----------------
  float wo0 = 0.0f, wo1 = 0.0f, wo2 = 0.0f, wo3 = 0.0f;
  float bo = -__builtin_inff();
  if (lane < TOUT) {
    wo0 = W_out[lane * THH + 0];
    wo1 = W_out[lane * THH + 1];
    wo2 = W_out[lane * THH + 2];
    wo3 = W_out[lane * THH + 3];
    bo  = b_out[lane];
  }
  for (int m = 0; m < 16; ++m) {
    float h0 = __shfl(hA.x, m, 32);        // row m units 0,1 live in lane m
    float h1 = __shfl(hA.y, m, 32);
    float h2 = __shfl(hA.x, m + 16, 32);   // units 2,3 live in lane m+16
    float h3 = __shfl(hA.y, m + 16, 32);
    float lg = fmaf(wo0, h0, fmaf(wo1, h1, fmaf(wo2, h2, fmaf(wo3, h3, bo))));
    float mx = lg;
    for (int s = 16; s; s >>= 1) mx = fmaxf(mx, __shfl_xor(mx, s, 32));
    float e = __builtin_amdgcn_exp2f((lg - mx) * LOG2E);  // lanes 30/31 -> 0
    float sum = e;
    for (int s = 16; s; s >>= 1) sum += __shfl_xor(sum, s, 32);
    if (lane < TOUT) out[(size_t)(b0 + m) * TOUT + lane] = e / sum;
  }
}

extern "C" void kernel_launch(void* const* d_in, const int* in_sizes, int n_in,
                              void* d_out, int out_size, void* d_ws, size_t ws_size,
                              hipStream_t stream) {
  (void)in_sizes; (void)n_in; (void)d_ws; (void)ws_size; (void)out_size;
  const float* X    = (const float*)d_in[0];
  const float* W_ih = (const float*)d_in[1];
  const float* W_hh = (const float*)d_in[2];
  const float* b_ih = (const float*)d_in[3];
  const float* b_hh = (const float*)d_in[4];
  const float* W_out= (const float*)d_in[5];
  const float* b_out= (const float*)d_in[6];
  float* out = (float*)d_out;

  // 4096 rows / 16 rows-per-wave = 256 single-wave workgroups -> spreads
  // across WGPs; no intra-block barriers needed.
  lstm_wmma_scan<<<4096 / 16, 32, 0, stream>>>(X, W_ih, W_hh, b_ih, b_hh,
                                               W_out, b_out, out);
}